// Attention_72473278152906
// MI455X (gfx1250) — compile-verified
//
#include <hip/hip_runtime.h>
#include <math.h>

// Paged GQA decode attention for MI455X (gfx1250, wave32).
// One workgroup per (batch, kv_head); 4 waves flash-decode disjoint 16-pos
// tiles with online softmax; partials merged in LDS.
// Matrix ops: V_WMMA_F32_16X16X4_F32 (fp32 to match the fp32 reference).
// K stream: double-buffered GLOBAL_LOAD_ASYNC_TO_LDS_B128 (one 512B K row per
// instruction) with XOR-swizzled LDS layout for conflict-free DS reads.

#define B_    32
#define H_    32
#define KVH_  8
#define D_    128
#define BS_   16
#define MAXB_ 256
#define G_    4                      // H / KVH  (queries per kv head)
#define SCALE_ 0.08838834764831845f

typedef float v2f __attribute__((ext_vector_type(2)));
typedef float v8f __attribute__((ext_vector_type(8)));
typedef int   v4i __attribute__((vector_size(4 * sizeof(int))));

#if __has_builtin(__builtin_amdgcn_global_load_async_to_lds_b128) && \
    __has_builtin(__builtin_amdgcn_s_wait_asynccnt)
#define USE_ASYNC 1
#else
#define USE_ASYNC 0
#endif

#define AS1 __attribute__((address_space(1)))
#define AS3 __attribute__((address_space(3)))

__global__ __launch_bounds__(128)
void paged_gqa_decode_kernel(const float* __restrict__ q,
                             const float* __restrict__ knew,
                             const float* __restrict__ vnew,
                             const float* __restrict__ kcache,
                             const float* __restrict__ vcache,
                             const int*   __restrict__ block_tables,
                             const int*   __restrict__ context_lens,
                             float*       __restrict__ out)
{
    const int bk  = blockIdx.x;            // b * KVH + kvh
    const int b   = bk / KVH_;
    const int kvh = bk % KVH_;
    const int h0  = kvh * G_;

    const int tid  = threadIdx.x;
    const int wid  = tid >> 5;             // wave id (0..3)
    const int lane = tid & 31;
    const int l16  = lane & 15;
    const int hi   = (lane >> 4) & 1;      // which 16-lane half

    const int ctx      = context_lens[b];
    const int last     = ctx - 1;          // fresh k/v token lives here
    const int numTiles = (ctx + BS_ - 1) / BS_;

#if USE_ASYNC
    // K staging: [wave][double-buf][16 rows x 128 floats], 16B segments XOR-
    // swizzled by row so GEMM1 DS reads are bank-conflict-free. The combine
    // scratch aliases the staging space (protected by __syncthreads below).
    __shared__ union {
        float kst[4][2][BS_ * D_];                      // 64 KB
        struct { float acc[4][G_][D_]; float m[4][G_]; float l[4][G_]; } fin;
    } sm;
#define SM_ACC sm.fin.acc
#define SM_M   sm.fin.m
#define SM_L   sm.fin.l
#else
    __shared__ float s_acc[4][G_][D_];
    __shared__ float s_m[4][G_];
    __shared__ float s_l[4][G_];
#define SM_ACC s_acc
#define SM_M   s_m
#define SM_L   s_l
#endif

    // ---- Preload Q as the B-operand of S^T = K * Q^T (scaled once). ----
    // B chunk kk (K-dim = dims 4kk..4kk+3): VGPR0 lanes0-15 -> d=4kk, q=lane;
    // lanes16-31 -> d=4kk+2, q=lane-16; VGPR1 = d+1. Columns q>=4 are zero.
    v2f bq[32];
    {
        const float* qrow = q + ((size_t)(b * H_ + h0 + (l16 < G_ ? l16 : 0))) * D_;
        #pragma unroll
        for (int kk = 0; kk < 32; ++kk) {
            if (l16 < G_) {
                const float2 t = *(const float2*)(qrow + 4 * kk + 2 * hi);
                bq[kk] = (v2f){t.x * SCALE_, t.y * SCALE_};
            } else {
                bq[kk] = (v2f){0.f, 0.f};
            }
        }
    }

    // Per-lane online-softmax state (lane owns query column q = l16).
    float m_run = -INFINITY;
    float l_run = 0.f;
    v8f acc[8];                            // O^T tiles: 8 d-tiles of 16
    #pragma unroll
    for (int t = 0; t < 8; ++t) acc[t] = (v8f){0, 0, 0, 0, 0, 0, 0, 0};

    const size_t kvrow  = (size_t)KVH_ * D_;          // stride between positions
    const float* kfresh = knew + ((size_t)(b * KVH_ + kvh)) * D_;
    const float* vfresh = vnew + ((size_t)(b * KVH_ + kvh)) * D_;

#if USE_ASYNC
    // Stage one 16-position K tile: 16 async B128 instructions, each moving a
    // full 512B K row (32 lanes x 16B contiguous) into this wave's LDS buffer.
    auto stage_k = [&](int tile, int buf) {
        const int bt = block_tables[b * MAXB_ + tile];
        float* dstbase = &sm.kst[wid][buf][0];
        #pragma unroll
        for (int i = 0; i < BS_; ++i) {
            const int gp = tile * BS_ + i;
            const float* src = (gp == last)
                ? (kfresh + lane * 4)
                : (kcache + (((size_t)bt * BS_ + i) * KVH_ + kvh) * D_ + lane * 4);
            float* dst = dstbase + i * D_ + ((lane ^ i) << 2);  // seg' = seg ^ row
            __builtin_amdgcn_global_load_async_to_lds_b128(
                (AS1 v4i*)src, (AS3 v4i*)dst, 0, 0);
        }
    };
    int buf = 0;
    if (wid < numTiles) stage_k(wid, 0);
#endif

    for (int tile = wid; tile < numTiles; tile += 4) {
        const int bt = block_tables[b * MAXB_ + tile];
        const float* vbase = vcache + (((size_t)bt * BS_) * KVH_ + kvh) * D_;

        // ---- GEMM1: S^T(16pos x 16q) = K(16pos x 128) * Q^T(128 x 16q) ----
        v8f S = (v8f){0, 0, 0, 0, 0, 0, 0, 0};
#if USE_ASYNC
        const int nxt = tile + 4;
        if (nxt < numTiles) {
            stage_k(nxt, buf ^ 1);                       // prefetch next tile
            __builtin_amdgcn_s_wait_asynccnt(16);        // current buffer done
        } else {
            __builtin_amdgcn_s_wait_asynccnt(0);
        }
        const float* kst_w = &sm.kst[wid][buf][0];
        #pragma unroll
        for (int kk = 0; kk < 32; ++kk) {
            // lane row = pos l16; cols d = 4kk + 2*hi, +1 (swizzled DS read)
            const float2 t = *(const float2*)(kst_w + l16 * D_ +
                                              (((unsigned)(kk ^ l16)) << 2) + 2 * hi);
            const v2f a = (v2f){t.x, t.y};
            S = __builtin_amdgcn_wmma_f32_16x16x4_f32(false, a, false, bq[kk],
                                                      (short)0, S, false, false);
        }
        buf ^= 1;
#else
        const float* kbase = kcache + (((size_t)bt * BS_) * KVH_ + kvh) * D_;
        const int    kpos  = tile * BS_ + l16;
        const float* krow  = (kpos == last) ? kfresh : (kbase + (size_t)l16 * kvrow);
        #pragma unroll
        for (int kk = 0; kk < 32; ++kk) {
            const float2 t = *(const float2*)(krow + 4 * kk + 2 * hi);
            const v2f a = (v2f){t.x, t.y};
            S = __builtin_amdgcn_wmma_f32_16x16x4_f32(false, a, false, bq[kk],
                                                      (short)0, S, false, false);
        }
#endif

        // ---- mask + online softmax (row = position, col = query = l16) ----
        float p[8];
        float tmax = -INFINITY;
        #pragma unroll
        for (int r = 0; r < 8; ++r) {
            const int gp = tile * BS_ + r + hi * 8;
            const float s = (gp < ctx) ? S[r] : -INFINITY;
            p[r] = s;
            tmax = fmaxf(tmax, s);
        }
        tmax = fmaxf(tmax, __shfl_xor(tmax, 16));      // combine halves
        const float m_new = fmaxf(m_run, tmax);
        const float corr  = __expf(m_run - m_new);
        float tsum = 0.f;
        #pragma unroll
        for (int r = 0; r < 8; ++r) {
            p[r] = __expf(p[r] - m_new);
            tsum += p[r];
        }
        tsum += __shfl_xor(tsum, 16);
        l_run = l_run * corr + tsum;
        m_run = m_new;
        #pragma unroll
        for (int t = 0; t < 8; ++t)
            #pragma unroll
            for (int e = 0; e < 8; ++e) acc[t][e] *= corr;

        // ---- GEMM2: O^T(16d x 16q) += V^T(16d x 16pos) * W(16pos x 16q) ----
        #pragma unroll
        for (int c = 0; c < 4; ++c) {
            // Rebuild W chunk (B operand) from S^T lane layout via shuffles:
            // value p[pos][q] lives at VGPR pos%8, lane (pos/8)*16 + q.
            const int pLo0 = 4 * c,     pLo1 = 4 * c + 1;
            const int pHi0 = 4 * c + 2, pHi1 = 4 * c + 3;
            const float b0lo = __shfl(p[pLo0 & 7], ((pLo0 >> 3) << 4) + l16);
            const float b0hi = __shfl(p[pHi0 & 7], ((pHi0 >> 3) << 4) + l16);
            const float b1lo = __shfl(p[pLo1 & 7], ((pLo1 >> 3) << 4) + l16);
            const float b1hi = __shfl(p[pHi1 & 7], ((pHi1 >> 3) << 4) + l16);
            const v2f bw = (v2f){hi ? b0hi : b0lo, hi ? b1hi : b1lo};

            const int pA = 4 * c + 2 * hi;
            const int gA = tile * BS_ + pA;
            const float* vr0 = (gA     == last) ? vfresh : (vbase + (size_t)pA       * kvrow);
            const float* vr1 = (gA + 1 == last) ? vfresh : (vbase + (size_t)(pA + 1) * kvrow);
            #pragma unroll
            for (int dt = 0; dt < 8; ++dt) {
                const int d = dt * 16 + l16;
                const v2f a = (v2f){vr0[d], vr1[d]};   // V^T chunk: row d, cols pA, pA+1
                acc[dt] = __builtin_amdgcn_wmma_f32_16x16x4_f32(false, a, false, bw,
                                                                (short)0, acc[dt], false, false);
            }
        }
    }

    // ---- merge the 4 wave partials (flash-decoding combine) ----
    __syncthreads();   // all waves done with K staging LDS (union aliasing)

    if (l16 < G_) {
        if (!hi) { SM_M[wid][l16] = m_run; SM_L[wid][l16] = l_run; }
        #pragma unroll
        for (int dt = 0; dt < 8; ++dt)
            #pragma unroll
            for (int r = 0; r < 8; ++r)
                SM_ACC[wid][l16][dt * 16 + r + hi * 8] = acc[dt][r];
    }
    __syncthreads();

    for (int idx = tid; idx < G_ * D_; idx += 128) {
        const int qq = idx / D_;
        const int d  = idx % D_;
        float M = -INFINITY;
        #pragma unroll
        for (int w = 0; w < 4; ++w) M = fmaxf(M, SM_M[w][qq]);
        float Lsum = 0.f, Osum = 0.f;
        #pragma unroll
        for (int w = 0; w < 4; ++w) {
            const float f = __expf(SM_M[w][qq] - M);
            Lsum += SM_L[w][qq] * f;
            Osum += SM_ACC[w][qq][d] * f;
        }
        out[((size_t)(b * H_ + h0 + qq)) * D_ + d] = Osum / Lsum;
    }
}

extern "C" void kernel_launch(void* const* d_in, const int* in_sizes, int n_in,
                              void* d_out, int out_size, void* d_ws, size_t ws_size,
                              hipStream_t stream) {
    (void)in_sizes; (void)n_in; (void)d_ws; (void)ws_size; (void)out_size;
    const float* q        = (const float*)d_in[0];
    const float* knew     = (const float*)d_in[1];
    const float* vnew     = (const float*)d_in[2];
    const float* kcache   = (const float*)d_in[3];
    const float* vcache   = (const float*)d_in[4];
    const int*   btables  = (const int*)d_in[5];
    const int*   ctxlens  = (const int*)d_in[6];
    // d_in[7] (slot_mapping) is implied by block_tables/context_lens; unused.
    float* out = (float*)d_out;

    dim3 grid(B_ * KVH_);
    dim3 block(128);
    paged_gqa_decode_kernel<<<grid, block, 0, stream>>>(
        q, knew, vnew, kcache, vcache, btables, ctxlens, out);
}